// GIN_43559558316084
// MI455X (gfx1250) — compile-verified
//
#include <hip/hip_runtime.h>
#include <hip/hip_bf16.h>
#include <math.h>

#define DN_NODES   100000
#define DD_IN      128
#define DD_H       64
#define DN_GRAPHS  256
#define DBN_EPS    1e-5f
#define PAD        4          // LDS row padding (floats): keeps 16B alignment,
                              // gives 4-bank row shift -> conflict-free column reads

typedef float v2f __attribute__((ext_vector_type(2)));
typedef float v8f __attribute__((ext_vector_type(8)));

// ---------------------------------------------------------------- zero fill (b128)
__global__ void zero4_kernel(float4* __restrict__ p, long n4) {
    long i = (long)blockIdx.x * blockDim.x + threadIdx.x;
    if (i < n4) p[i] = make_float4(0.f, 0.f, 0.f, 0.f);
}

// ---------------------------------------------------------------- edge scatter-add
// float4 per thread: thread t -> edge e = t/(D/4), quad c4 = t%(D/4).
// D=128 -> 32 threads/edge (src/dst wave-uniform); D=64 -> 16 threads/edge.
// global_load_b128 gather + 4x global_atomic_add_f32 scatter.
template<int D>
__global__ void scatter_add_kernel(const float4* __restrict__ X4,
                                   const int* __restrict__ src,
                                   const int* __restrict__ dst,
                                   float* __restrict__ agg,
                                   int n_edges) {
    constexpr int Q = D / 4;
    long t = (long)blockIdx.x * blockDim.x + threadIdx.x;
    int e  = (int)(t / Q);
    int c4 = (int)(t % Q);
    if (e < n_edges) {
        int s = src[e];
        int d = dst[e];
        float4 v = X4[(long)s * Q + c4];
        float* a = &agg[(long)d * D + c4 * 4];
        unsafeAtomicAdd(a + 0, v.x);
        unsafeAtomicAdd(a + 1, v.y);
        unsafeAtomicAdd(a + 2, v.z);
        unsafeAtomicAdd(a + 3, v.w);
    }
}

// ---------------------------------------------------------------- fused GIN layer MLP
// H = relu( relu(BN((X+Agg)@W1 + b1)) @ W2 + b2 )
// block = 256 threads (8 waves); 32 nodes x 64 output features per block.
// Waves laid out 2 (M-tiles) x 4 (N-tiles), each owning a 16x16 WMMA tile.
// WMMA f32 16x16x4 layouts (ISA 7.12.2):
//   A 16x4 : lane<16 -> M=lane,   vgpr j holds K=j   ; lane>=16 -> M=lane-16, K=j+2
//   B 4x16 : lane<16 -> N=lane,   vgpr j holds K=j   ; lane>=16 -> N=lane-16, K=j+2
//   C 16x16: vgpr j -> M = j + (lane<16 ? 0 : 8), N = lane&15
template<int DIN>
__global__ void __launch_bounds__(256)
gin_mlp_wmma_kernel(const float* __restrict__ X,  const float* __restrict__ Agg,
                    const float* __restrict__ W1, const float* __restrict__ B1,
                    const float* __restrict__ G,  const float* __restrict__ BT,
                    const float* __restrict__ RM, const float* __restrict__ RV,
                    const float* __restrict__ W2, const float* __restrict__ B2,
                    float* __restrict__ H, int n_nodes) {
    constexpr int LDX = DIN + PAD;
    constexpr int LDH = DD_H + PAD;
    __shared__ float sX[32 * LDX];
    __shared__ float sH[32 * LDH];

    const int tid   = threadIdx.x;
    const int node0 = blockIdx.x * 32;

    // stage (x + agg) into LDS: float4 loads + b128 LDS stores, coalesced
    {
        constexpr int Q = DIN / 4;
        const float4* X4 = (const float4*)X;
        const float4* A4 = (const float4*)Agg;
        for (int idx = tid; idx < 32 * Q; idx += 256) {
            int r = idx / Q, c4 = idx % Q;
            int node = node0 + r;
            float4 v = make_float4(0.f, 0.f, 0.f, 0.f);
            if (node < n_nodes) {
                float4 xv = X4[(long)node * Q + c4];
                float4 av = A4[(long)node * Q + c4];
                v = make_float4(xv.x + av.x, xv.y + av.y, xv.z + av.z, xv.w + av.w);
            }
            *(float4*)&sX[r * LDX + c4 * 4] = v;   // row stride 16B-aligned
        }
    }
    __syncthreads();

    const int lane  = tid & 31;
    const int wave  = tid >> 5;
    const int baseM = (wave >> 2) * 16;   // 0 or 16
    const int baseN = (wave & 3) * 16;    // 0,16,32,48
    const int l15   = lane & 15;
    const int hi    = lane >> 4;          // 0 or 1
    const int col   = baseN + l15;        // output column owned by this lane
    const int arow  = baseM + l15;        // A-fragment row owned by this lane

    // ---------------- GEMM1: (X+Agg) @ W1 ----------------
    v8f acc = {0.f, 0.f, 0.f, 0.f, 0.f, 0.f, 0.f, 0.f};
    #pragma unroll 4
    for (int k0 = 0; k0 < DIN; k0 += 4) {
        int kl = k0 + (hi << 1);
        v2f a, b;
        a[0] = sX[arow * LDX + kl];
        a[1] = sX[arow * LDX + kl + 1];
        b[0] = W1[kl * DD_H + col];
        b[1] = W1[(kl + 1) * DD_H + col];
        acc = __builtin_amdgcn_wmma_f32_16x16x4_f32(false, a, false, b,
                                                    (short)0, acc, false, false);
    }

    // bias + BatchNorm(eval) + ReLU, write intermediate to LDS
    float scale = G[col] * rsqrtf(RV[col] + DBN_EPS);
    float shift = BT[col] - RM[col] * scale;
    float bias1 = B1[col];
    #pragma unroll
    for (int j = 0; j < 8; j++) {
        int r = baseM + j + hi * 8;
        float h = (acc[j] + bias1) * scale + shift;
        sH[r * LDH + col] = fmaxf(h, 0.0f);
    }
    __syncthreads();

    // ---------------- GEMM2: h @ W2 ----------------
    v8f acc2 = {0.f, 0.f, 0.f, 0.f, 0.f, 0.f, 0.f, 0.f};
    #pragma unroll 4
    for (int k0 = 0; k0 < DD_H; k0 += 4) {
        int kl = k0 + (hi << 1);
        v2f a, b;
        a[0] = sH[arow * LDH + kl];
        a[1] = sH[arow * LDH + kl + 1];
        b[0] = W2[kl * DD_H + col];
        b[1] = W2[(kl + 1) * DD_H + col];
        acc2 = __builtin_amdgcn_wmma_f32_16x16x4_f32(false, a, false, b,
                                                     (short)0, acc2, false, false);
    }

    float bias2 = B2[col];
    #pragma unroll
    for (int j = 0; j < 8; j++) {
        int r = baseM + j + hi * 8;
        int node = node0 + r;
        if (node < n_nodes)
            H[(long)node * DD_H + col] = fmaxf(acc2[j] + bias2, 0.0f);
    }
}

// ---------------------------------------------------------------- graph pooling
// float4 per thread: thread t -> node i = t/16, quad f4 = t%16.
__global__ void pool_kernel(const float4* __restrict__ H4,
                            const int* __restrict__ batch,
                            float* __restrict__ P, int n_nodes, int off) {
    long t = (long)blockIdx.x * blockDim.x + threadIdx.x;
    int i  = (int)(t >> 4);
    int f4 = (int)(t & 15);
    if (i < n_nodes) {
        int g = batch[i];
        float4 v = H4[(long)i * 16 + f4];
        float* p = &P[(long)g * (3 * DD_H) + off + f4 * 4];
        unsafeAtomicAdd(p + 0, v.x);
        unsafeAtomicAdd(p + 1, v.y);
        unsafeAtomicAdd(p + 2, v.z);
        unsafeAtomicAdd(p + 3, v.w);
    }
}

// ---------------------------------------------------------------- classifier head
// per graph: relu(P@lin1_w + b) @ lin2_w + b2 -> log_softmax (2 classes)
__global__ void __launch_bounds__(64)
classify_kernel(const float* __restrict__ P,
                const float* __restrict__ W1, const float* __restrict__ B1,
                const float* __restrict__ W2, const float* __restrict__ B2,
                float* __restrict__ Out) {
    const int g = blockIdx.x;
    const int t = threadIdx.x;   // 64 threads
    __shared__ float sp[3 * DD_H];
    __shared__ float sh[64];

    for (int i = t; i < 3 * DD_H; i += 64) sp[i] = P[(long)g * (3 * DD_H) + i];
    __syncthreads();

    float acc = B1[t];
    for (int k = 0; k < 3 * DD_H; k++) acc += sp[k] * W1[k * 64 + t];
    sh[t] = fmaxf(acc, 0.0f);
    __syncthreads();

    if (t == 0) {
        float z0 = B2[0], z1 = B2[1];
        for (int k = 0; k < 64; k++) {
            z0 += sh[k] * W2[k * 2 + 0];
            z1 += sh[k] * W2[k * 2 + 1];
        }
        float m   = fmaxf(z0, z1);
        float lse = m + logf(expf(z0 - m) + expf(z1 - m));
        Out[g * 2 + 0] = z0 - lse;
        Out[g * 2 + 1] = z1 - lse;
    }
}

// ---------------------------------------------------------------- launcher
extern "C" void kernel_launch(void* const* d_in, const int* in_sizes, int n_in,
                              void* d_out, int out_size, void* d_ws, size_t ws_size,
                              hipStream_t stream) {
    const float* x      = (const float*)d_in[0];
    const int*   ei     = (const int*)d_in[1];
    const int*   batch  = (const int*)d_in[2];
    // d_in[3] = num_graphs (device scalar) -> compile-time 256
    const float* w11 = (const float*)d_in[4],  *b11 = (const float*)d_in[5];
    const float* g1  = (const float*)d_in[6],  *bt1 = (const float*)d_in[7];
    const float* rm1 = (const float*)d_in[8],  *rv1 = (const float*)d_in[9];
    const float* w12 = (const float*)d_in[10], *b12 = (const float*)d_in[11];
    const float* w21 = (const float*)d_in[12], *b21 = (const float*)d_in[13];
    const float* g2  = (const float*)d_in[14], *bt2 = (const float*)d_in[15];
    const float* rm2 = (const float*)d_in[16], *rv2 = (const float*)d_in[17];
    const float* w22 = (const float*)d_in[18], *b22 = (const float*)d_in[19];
    const float* w31 = (const float*)d_in[20], *b31 = (const float*)d_in[21];
    const float* g3  = (const float*)d_in[22], *bt3 = (const float*)d_in[23];
    const float* rm3 = (const float*)d_in[24], *rv3 = (const float*)d_in[25];
    const float* w32 = (const float*)d_in[26], *b32 = (const float*)d_in[27];
    const float* l1w = (const float*)d_in[28], *l1b = (const float*)d_in[29];
    const float* l2w = (const float*)d_in[30], *l2b = (const float*)d_in[31];
    float* out = (float*)d_out;

    const int n_nodes = in_sizes[2];
    const int n_edges = in_sizes[1] / 2;
    const int* src = ei;
    const int* dst = ei + n_edges;

    // workspace layout (floats, all offsets 16B-aligned)
    float* agg = (float*)d_ws;                               // n_nodes*128 (reused as n_nodes*64)
    float* h1  = agg + (long)n_nodes * DD_IN;                // n_nodes*64 each
    float* h2  = h1  + (long)n_nodes * DD_H;
    float* h3  = h2  + (long)n_nodes * DD_H;
    float* P   = h3  + (long)n_nodes * DD_H;                 // 256*192

    const int blk = 256;
    long n;

    // ---------- layer 1 (DIN=128) ----------
    n = (long)n_nodes * DD_IN / 4;
    zero4_kernel<<<(int)((n + blk - 1) / blk), blk, 0, stream>>>((float4*)agg, n);
    n = (long)n_edges * (DD_IN / 4);
    scatter_add_kernel<DD_IN><<<(int)((n + blk - 1) / blk), blk, 0, stream>>>(
        (const float4*)x, src, dst, agg, n_edges);
    gin_mlp_wmma_kernel<DD_IN><<<(n_nodes + 31) / 32, blk, 0, stream>>>(
        x, agg, w11, b11, g1, bt1, rm1, rv1, w12, b12, h1, n_nodes);

    // ---------- layer 2 (DIN=64) ----------
    n = (long)n_nodes * DD_H / 4;
    zero4_kernel<<<(int)((n + blk - 1) / blk), blk, 0, stream>>>((float4*)agg, n);
    n = (long)n_edges * (DD_H / 4);
    scatter_add_kernel<DD_H><<<(int)((n + blk - 1) / blk), blk, 0, stream>>>(
        (const float4*)h1, src, dst, agg, n_edges);
    gin_mlp_wmma_kernel<DD_H><<<(n_nodes + 31) / 32, blk, 0, stream>>>(
        h1, agg, w21, b21, g2, bt2, rm2, rv2, w22, b22, h2, n_nodes);

    // ---------- layer 3 (DIN=64) ----------
    n = (long)n_nodes * DD_H / 4;
    zero4_kernel<<<(int)((n + blk - 1) / blk), blk, 0, stream>>>((float4*)agg, n);
    n = (long)n_edges * (DD_H / 4);
    scatter_add_kernel<DD_H><<<(int)((n + blk - 1) / blk), blk, 0, stream>>>(
        (const float4*)h2, src, dst, agg, n_edges);
    gin_mlp_wmma_kernel<DD_H><<<(n_nodes + 31) / 32, blk, 0, stream>>>(
        h2, agg, w31, b31, g3, bt3, rm3, rv3, w32, b32, h3, n_nodes);

    // ---------- pooling ----------
    n = (long)DN_GRAPHS * 3 * DD_H / 4;
    zero4_kernel<<<(int)((n + blk - 1) / blk), blk, 0, stream>>>((float4*)P, n);
    n = (long)n_nodes * (DD_H / 4);
    pool_kernel<<<(int)((n + blk - 1) / blk), blk, 0, stream>>>(
        (const float4*)h1, batch, P, n_nodes, 0);
    pool_kernel<<<(int)((n + blk - 1) / blk), blk, 0, stream>>>(
        (const float4*)h2, batch, P, n_nodes, DD_H);
    pool_kernel<<<(int)((n + blk - 1) / blk), blk, 0, stream>>>(
        (const float4*)h3, batch, P, n_nodes, 2 * DD_H);

    // ---------- classifier head ----------
    classify_kernel<<<DN_GRAPHS, 64, 0, stream>>>(P, l1w, l1b, l2w, l2b, out);
}